// Essential2PoseCandidates_28132035789008
// MI455X (gfx1250) — compile-verified
//
#include <hip/hip_runtime.h>
#include <stdint.h>

// ---------------------------------------------------------------------------
// Essential matrix -> 4 pose candidates, one thread per 3x3 matrix.
// Global traffic handled entirely by the CDNA5 Tensor Data Mover (TDM):
//   per 256-thread block:  1x tensor_load_to_lds    (2304 dwords of E)
//                          2x tensor_store_from_lds (9216 dwords R, 3072 dwords t)
// synchronized with s_wait_tensorcnt + workgroup barriers.
// Compute: symmetric cyclic-Jacobi eigensolver on E^T E using HW rcp/rsq/sqrt
// (single-instruction transcendentals) to stay at the HBM roofline.
// ---------------------------------------------------------------------------

typedef uint32_t v4u __attribute__((ext_vector_type(4)));
typedef int      v8i __attribute__((ext_vector_type(8)));
typedef int      v4i __attribute__((ext_vector_type(4)));

// Build a 1-D TDM descriptor (n dwords, n < 65536) and issue load or store.
// Field packing per CDNA5 ISA §8 (groups 0..3), data_size = 4 bytes.
// clang-23 builtin form: (v4u g0, v8i g1, v4i g2, v4i g3, v8i reserved, i32 cpol)
__device__ __forceinline__ void tdm_issue(bool is_store, uint32_t lds_byte,
                                          uint64_t gaddr, uint32_t ndw) {
  uint32_t g0[4], g1[8], g2[4], g3[4];
  // group0: count=1 | lds_addr | global_addr[56:0] | type=2
  g0[0] = 1u;
  g0[1] = lds_byte;
  g0[2] = (uint32_t)gaddr;
  g0[3] = (uint32_t)((gaddr >> 32) & 0x01FFFFFFu) | (2u << 30);
  // group1: data_size=2(4B); tensor_dim0=n; tensor_dim1=1; tile_dim0=n;
  //         tile_dim1=1; tile_dim2=1; strides=n
  g1[0] = 2u << 16;                     // [17:16] data_size
  g1[1] = (ndw & 0xFFFFu) << 16;        // tensor_dim0 lo16  -> bits[63:48]
  g1[2] = (ndw >> 16) | (1u << 16);     // tensor_dim0 hi16, tensor_dim1=1
  g1[3] = (ndw & 0xFFFFu) << 16;        // tile_dim0         -> bits[127:112]
  g1[4] = 1u | (1u << 16);              // tile_dim1=1, tile_dim2=1
  g1[5] = ndw;                          // tensor_dim0_stride lo32
  g1[6] = (ndw & 0xFFFFu) << 16;        // stride0 hi16=0 | tensor_dim1_stride lo16
  g1[7] = 0u;                           // tensor_dim1_stride hi32
  // group2: tensor_dim2=1, tensor_dim3=1, tensor_dim2_stride=n, tile_dim3=1
  g2[0] = 1u; g2[1] = 1u; g2[2] = ndw; g2[3] = 1u << 16;
  // group3: tensor_dim3_stride=n, tensor_dim4=1, tile_dim4=1
  g3[0] = ndw; g3[1] = 1u << 16; g3[2] = 1u << 16; g3[3] = 0u;

  v8i zero8 = (v8i)0;  // reserved operand in the 6-arg builtin form

  if (is_store)
    __builtin_amdgcn_tensor_store_from_lds(*(v4u*)g0, *(v8i*)g1,
                                           *(v4i*)g2, *(v4i*)g3, zero8, 0);
  else
    __builtin_amdgcn_tensor_load_to_lds(*(v4u*)g0, *(v8i*)g1,
                                        *(v4i*)g2, *(v4i*)g3, zero8, 0);
}

// One cyclic-Jacobi rotation on the symmetric 3x3 (6 unique entries),
// zeroing a_pq.  P,Q select the pivot pair; (arp, arq) are the two entries
// coupling the third index r to p and q.  V accumulates V = V*J.
// Division-free tangent:  t = 2*apq*sign(h) / (|h| + sqrt(h^2 + 4*apq^2)),
// with HW-approximate rcp/rsq/sqrt (1 instruction each).
template <int P, int Q>
__device__ __forceinline__ void jrot(float& app, float& aqq, float& apq,
                                     float& arp, float& arq, float V[3][3]) {
  float h   = aqq - app;
  float p2  = apq + apq;
  float mag = __builtin_amdgcn_sqrtf(h * h + p2 * p2);
  float den = fabsf(h) + mag;
  float t   = p2 * copysignf(1.0f, h) * __builtin_amdgcn_rcpf(den + 1e-37f);
  float c   = __builtin_amdgcn_rsqf(t * t + 1.0f);
  float s   = t * c;

  float ta = t * apq;          // symmetric diagonal update
  app -= ta;
  aqq += ta;
  apq  = 0.0f;
  float rp = arp, rq = arq;    // the single off-pair (k = r)
  arp = c * rp - s * rq;
  arq = s * rp + c * rq;
#pragma unroll
  for (int k = 0; k < 3; ++k) {  // V = V*J
    float vkp = V[k][P], vkq = V[k][Q];
    V[k][P] = c * vkp - s * vkq;
    V[k][Q] = s * vkp + c * vkq;
  }
}

__global__ __launch_bounds__(256) void
pose_candidates_kernel(const float* __restrict__ Ein, float* __restrict__ out, int Btot) {
  // LDS staging: [0, 9216) floats = R candidates, [9216, 12288) = t candidates.
  // Input (2304 floats) is staged into the front of the same buffer first.
  __shared__ __align__(16) float lds[12288];

  const int      tid      = threadIdx.x;
  const int      blk      = blockIdx.x;
  const uint32_t lds_base = (uint32_t)(uintptr_t)(&lds[0]);

  // ---- Phase A: TDM load of this block's 256 matrices (wave 0 only) ----
  if (__builtin_amdgcn_readfirstlane((int)threadIdx.x) == 0) {
    uint64_t src = (uint64_t)(uintptr_t)Ein + (uint64_t)blk * 9216u; // bytes
    tdm_issue(false, lds_base, src, 2304u);
    __builtin_amdgcn_s_wait_tensorcnt(0);
  }
  asm volatile("" ::: "memory");
  __syncthreads();

  // ---- Phase B: read my matrix (bank-conflict-free: stride 9, gcd(9,64)=1) ----
  float e[3][3];
#pragma unroll
  for (int i = 0; i < 3; ++i)
#pragma unroll
    for (int j = 0; j < 3; ++j) e[i][j] = lds[9 * tid + 3 * i + j];
  __syncthreads();  // everyone done reading before we overwrite LDS

  // ---- Phase C: eigen-decomposition of A = E^T E (6 unique entries) ----
  float a00 = e[0][0]*e[0][0] + e[1][0]*e[1][0] + e[2][0]*e[2][0];
  float a11 = e[0][1]*e[0][1] + e[1][1]*e[1][1] + e[2][1]*e[2][1];
  float a22 = e[0][2]*e[0][2] + e[1][2]*e[1][2] + e[2][2]*e[2][2];
  float a01 = e[0][0]*e[0][1] + e[1][0]*e[1][1] + e[2][0]*e[2][1];
  float a02 = e[0][0]*e[0][2] + e[1][0]*e[1][2] + e[2][0]*e[2][2];
  float a12 = e[0][1]*e[0][2] + e[1][1]*e[1][2] + e[2][1]*e[2][2];

  float V[3][3] = {{1.f, 0.f, 0.f}, {0.f, 1.f, 0.f}, {0.f, 0.f, 1.f}};
#pragma unroll
  for (int sweep = 0; sweep < 5; ++sweep) {
    jrot<0, 1>(a00, a11, a01, a02, a12, V);  // r=2: (a20,a21)=(a02,a12)
    jrot<0, 2>(a00, a22, a02, a01, a12, V);  // r=1: (a10,a12)=(a01,a12)
    jrot<1, 2>(a11, a22, a12, a01, a02, V);  // r=0: (a01,a02)
  }

  // Sort eigenvalues descending, permuting V columns (3-element network).
  float l0 = a00, l1 = a11, l2 = a22;
  if (l0 < l1) { float t = l0; l0 = l1; l1 = t;
#pragma unroll
    for (int k = 0; k < 3; ++k) { float v = V[k][0]; V[k][0] = V[k][1]; V[k][1] = v; } }
  if (l0 < l2) { float t = l0; l0 = l2; l2 = t;
#pragma unroll
    for (int k = 0; k < 3; ++k) { float v = V[k][0]; V[k][0] = V[k][2]; V[k][2] = v; } }
  if (l1 < l2) { float t = l1; l1 = l2; l2 = t;
#pragma unroll
    for (int k = 0; k < 3; ++k) { float v = V[k][1]; V[k][1] = V[k][2]; V[k][2] = v; } }

  // U columns: u_c = normalize(E * v_c)  (sigma > 0, direction identical).
  float u[3][3];  // u[c][i] = component i of column c
#pragma unroll
  for (int c = 0; c < 3; ++c) {
    float x = e[0][0] * V[0][c] + e[0][1] * V[1][c] + e[0][2] * V[2][c];
    float y = e[1][0] * V[0][c] + e[1][1] * V[1][c] + e[1][2] * V[2][c];
    float z = e[2][0] * V[0][c] + e[2][1] * V[1][c] + e[2][2] * V[2][c];
    float rn = __builtin_amdgcn_rsqf(x * x + y * y + z * z + 1e-37f);
    u[c][0] = x * rn; u[c][1] = y * rn; u[c][2] = z * rn;
  }

  // Determinant sign fixes (whole-matrix flip, matching the reference).
  {
    float cx = u[1][1] * u[2][2] - u[1][2] * u[2][1];
    float cy = u[1][2] * u[2][0] - u[1][0] * u[2][2];
    float cz = u[1][0] * u[2][1] - u[1][1] * u[2][0];
    float du = u[0][0] * cx + u[0][1] * cy + u[0][2] * cz;
    float su = (du < 0.0f) ? -1.0f : 1.0f;
#pragma unroll
    for (int c = 0; c < 3; ++c)
#pragma unroll
      for (int i = 0; i < 3; ++i) u[c][i] *= su;
  }
  {
    float dv = V[0][0] * (V[1][1] * V[2][2] - V[1][2] * V[2][1])
             - V[0][1] * (V[1][0] * V[2][2] - V[1][2] * V[2][0])
             + V[0][2] * (V[1][0] * V[2][1] - V[1][1] * V[2][0]);
    float sv = (dv < 0.0f) ? -1.0f : 1.0f;
#pragma unroll
    for (int k = 0; k < 3; ++k)
#pragma unroll
      for (int j = 0; j < 3; ++j) V[k][j] *= sv;
  }

  // R1 = (U*W)*V, R2 = (U*W^T)*V with UW = [u1 | -u0 | u2], UW^T = [-u1 | u0 | u2].
  const int rb = 36 * tid;
#pragma unroll
  for (int i = 0; i < 3; ++i)
#pragma unroll
    for (int l = 0; l < 3; ++l) {
      float d  = u[1][i] * V[0][l] - u[0][i] * V[1][l];
      float cc = u[2][i] * V[2][l];
      float r1 = d + cc;
      float r2 = cc - d;
      lds[rb +      3 * i + l] = r1;   // candidate 0 = R1
      lds[rb +  9 + 3 * i + l] = r1;   // candidate 1 = R1
      lds[rb + 18 + 3 * i + l] = r2;   // candidate 2 = R2
      lds[rb + 27 + 3 * i + l] = r2;   // candidate 3 = R2
    }
  const int tb = 9216 + 12 * tid;
#pragma unroll
  for (int j = 0; j < 3; ++j) {
    float tv = u[2][j];                // t = U[:,2]
    lds[tb +     j] =  tv;
    lds[tb + 3 + j] = -tv;
    lds[tb + 6 + j] =  tv;
    lds[tb + 9 + j] = -tv;
  }

  asm volatile("" ::: "memory");
  __syncthreads();

  // ---- Phase D: TDM stores of R (36 KB) and t (12 KB) regions (wave 0) ----
  if (__builtin_amdgcn_readfirstlane((int)threadIdx.x) == 0) {
    uint64_t rDst = (uint64_t)(uintptr_t)out + (uint64_t)blk * 36864u;
    tdm_issue(true, lds_base, rDst, 9216u);
    uint64_t tDst = (uint64_t)(uintptr_t)out + (uint64_t)Btot * 144ull
                  + (uint64_t)blk * 12288u;
    tdm_issue(true, lds_base + 36864u, tDst, 3072u);
    __builtin_amdgcn_s_wait_tensorcnt(0);
  }
}

extern "C" void kernel_launch(void* const* d_in, const int* in_sizes, int n_in,
                              void* d_out, int out_size, void* d_ws, size_t ws_size,
                              hipStream_t stream) {
  const float* E   = (const float*)d_in[0];
  float*       out = (float*)d_out;
  const int Btot = in_sizes[0] / 9;     // 524288
  const int nblk = Btot / 256;          // exact: B is a multiple of 256
  pose_candidates_kernel<<<nblk, 256, 0, stream>>>(E, out, Btot);
}